// TemporalInterlace_82025285419382
// MI455X (gfx1250) — compile-verified
//
#include <hip/hip_runtime.h>
#include <math.h>

typedef float v4f __attribute__((ext_vector_type(4)));
typedef float v2f __attribute__((ext_vector_type(2)));
typedef float v8f __attribute__((ext_vector_type(8)));

#define HW    3136      // 56*56
#define HW4   784       // HW / 4
#define CTOT  256
#define NFOLD 64        // C / SHIFT_DIV
#define NB    8         // batch groups
#define NS    8         // segments
#define NOUT0 51380224  // 64*256*56*56

__device__ __forceinline__ int iclamp(int v, int lo, int hi) {
    return v < lo ? lo : (v > hi ? hi : v);
}

// ---------------------------------------------------------------------------
// Kernel 1: spatial mean-pool of the first 64 channels.
// One block per (n, fold): 4096 blocks x 256 threads.
// pooled layout: [b][fold][s]  (b = n/8, s = n%8)
// ---------------------------------------------------------------------------
__global__ void __launch_bounds__(256)
pool_kernel(const float* __restrict__ x, float* __restrict__ pooled) {
    int bid = blockIdx.x;           // 0..4095
    int n = bid >> 6;               // 0..63
    int f = bid & 63;               // fold channel
    const v4f* src = (const v4f*)(x + ((size_t)n * CTOT + f) * HW);
    float sum = 0.f;
    for (int i = threadIdx.x; i < HW4; i += 256) {
        v4f v = src[i];
        sum += v.x + v.y + v.z + v.w;
    }
    __shared__ float red[256];
    red[threadIdx.x] = sum;
    __syncthreads();
    for (int off = 128; off > 0; off >>= 1) {
        if (threadIdx.x < (unsigned)off) red[threadIdx.x] += red[threadIdx.x + off];
        __syncthreads();
    }
    if (threadIdx.x == 0) {
        int b = n >> 3, s = n & 7;
        pooled[(b * NFOLD + f) * NS + s] = red[0] * (1.0f / (float)HW);
    }
}

// ---------------------------------------------------------------------------
// Kernel 2: head — conv1d(offsets) -> fc1 (WMMA f32 16x16x4) -> fc2 ->
// sigmoid offsets, conv1d(weights) -> sigmoid weights, then precompute
// per-(b, part, s) gather taps {o0, o1} and coefficients {a0, a1}.
// Single wave32 block (EXEC all-ones for WMMA).
// ---------------------------------------------------------------------------
__global__ void __launch_bounds__(32)
head_kernel(const float* __restrict__ pooled,
            const float* __restrict__ cow,  const float* __restrict__ cob,
            const float* __restrict__ f1w,  const float* __restrict__ f1b,
            const float* __restrict__ f2w,  const float* __restrict__ f2b,
            const float* __restrict__ cww,  const float* __restrict__ cwb,
            float* __restrict__ out_off,    float* __restrict__ out_wt,
            int*   __restrict__ sampIdx,    float* __restrict__ sampA) {
    __shared__ float sP[NB * NFOLD * NS];   // 4096 floats pooled
    __shared__ float sXO[NB * NS];          // conv_off result  [b][s]
    __shared__ float sXO2[NB * NS];         // fc1+relu result  [b][j]
    __shared__ float sOff[NB * 2];          // negated offsets  [b][g]
    __shared__ float sWt[NB * NS * 2];      // weights          [b][s][g]

    const int lane = threadIdx.x;           // 0..31

    for (int i = lane; i < NB * NFOLD * NS; i += 32) sP[i] = pooled[i];
    __syncthreads();

    // ---- conv1d for offsets: xo[b][s] = bias + sum_f sum_k w[f][k]*P[b][f][s+k-1]
    for (int idx = lane; idx < NB * NS; idx += 32) {
        int b = idx >> 3, s = idx & 7;
        float acc = cob[0];
        for (int f = 0; f < NFOLD; ++f) {
            const float* Pr = &sP[(b * NFOLD + f) * NS];
            float pm = (s >= 1) ? Pr[s - 1] : 0.f;
            float pp = (s <= 6) ? Pr[s + 1] : 0.f;
            acc += cow[f * 3 + 0] * pm + cow[f * 3 + 1] * Pr[s] + cow[f * 3 + 2] * pp;
        }
        sXO[idx] = acc;
    }
    __syncthreads();

    // ---- fc1 as WMMA f32 16x16x4 (exact fp32), K=8 in two chunks of 4.
    // A[M=b][K=s] = xo;  B[K=s][N=j] = fc1_w[j][s];  C = A*B (8x8 live in 16x16).
    // A layout: lanes 0-15 -> K={0,1}, lanes 16-31 -> K={2,3} (VGPR0=K even, VGPR1=K odd).
    // B/C layout: N on lanes, rows on VGPRs, lane halves split rows.
    {
        int mn = lane & 15;            // A row M / B col N
        int kb = (lane >> 4) * 2;      // K base within 4-wide chunk
        v2f a0 = {0.f, 0.f}, a1 = {0.f, 0.f}, b0 = {0.f, 0.f}, b1 = {0.f, 0.f};
        if (mn < 8) {
            a0.x = sXO[mn * 8 + kb + 0];
            a0.y = sXO[mn * 8 + kb + 1];
            a1.x = sXO[mn * 8 + 4 + kb + 0];
            a1.y = sXO[mn * 8 + 4 + kb + 1];
            b0.x = f1w[mn * 8 + kb + 0];       // B[kb+0][mn]  = fc1_w[mn][kb+0]
            b0.y = f1w[mn * 8 + kb + 1];
            b1.x = f1w[mn * 8 + 4 + kb + 0];
            b1.y = f1w[mn * 8 + 4 + kb + 1];
        }
        v8f c = {0.f, 0.f, 0.f, 0.f, 0.f, 0.f, 0.f, 0.f};
        c = __builtin_amdgcn_wmma_f32_16x16x4_f32(false, a0, false, b0,
                                                  (short)0, c, false, false);
        c = __builtin_amdgcn_wmma_f32_16x16x4_f32(false, a1, false, b1,
                                                  (short)0, c, false, false);
        // C element (M=v for lanes<16, N=lane): lanes 0-7 / vgprs 0-7 hold all 8x8.
        if (lane < 8) {
            float bias = f1b[lane];
            for (int v = 0; v < 8; ++v)
                sXO2[v * 8 + lane] = fmaxf(0.f, c[v] + bias);
        }
    }
    __syncthreads();

    // ---- fc2 + sigmoid -> negated offsets (the reference returns -4*(sig-0.5))
    if (lane < 16) {
        int b = lane >> 1, g = lane & 1;
        float acc = f2b[g];
        for (int j = 0; j < 8; ++j) acc += sXO2[b * 8 + j] * f2w[g * 8 + j];
        float sig = 1.0f / (1.0f + expf(-acc));
        float offneg = -(4.0f * (sig - 0.5f));
        sOff[b * 2 + g] = offneg;
        out_off[b * 2 + g] = offneg;           // x_offset output (8,1,2)
    }
    __syncthreads();

    // ---- conv1d for weights + sigmoid: weight[b][s][g] = 2*sigmoid(xw[b][g][s])
    for (int idx = lane; idx < NB * 2 * NS; idx += 32) {
        int b = idx >> 4, g = (idx >> 3) & 1, s = idx & 7;
        float acc = cwb[g];
        for (int f = 0; f < NFOLD; ++f) {
            const float* Pr = &sP[(b * NFOLD + f) * NS];
            const float* W  = &cww[(g * NFOLD + f) * 3];
            float pm = (s >= 1) ? Pr[s - 1] : 0.f;
            float pp = (s <= 6) ? Pr[s + 1] : 0.f;
            acc += W[0] * pm + W[1] * Pr[s] + W[2] * pp;
        }
        float wt = 2.0f / (1.0f + expf(-acc));
        sWt[(b * 8 + s) * 2 + g] = wt;
        out_wt[(b * 8 + s) * 2 + g] = wt;      // x_weight output (8,8,2)
    }
    __syncthreads();

    // ---- sampler taps per (b, part, s); parts 0,1 use negated offset, 2,3 use +.
    for (int idx = lane; idx < NB * 4 * NS; idx += 32) {
        int b = idx >> 5, part = (idx >> 3) & 3, s = idx & 7, g = part & 1;
        float ob   = (part < 2) ? sOff[b * 2 + g] : -sOff[b * 2 + g];
        float offv = ob + (float)(s + 1);      // location = s+1
        int   fl   = (int)floorf(offv);
        int   o0   = iclamp(fl, 0, 9);
        int   o1   = iclamp(fl + 1, 0, 9);
        float w0   = 1.0f - (offv - (float)o0);
        float w1   = 1.0f - w0;
        float wt   = sWt[(b * 8 + s) * 2 + g];
        sampIdx[idx * 2 + 0] = o0;
        sampIdx[idx * 2 + 1] = o1;
        sampA[idx * 2 + 0]   = w0 * wt;
        sampA[idx * 2 + 1]   = w1 * wt;
    }
}

// ---------------------------------------------------------------------------
// Kernel 3: bulk traffic. One block per (n, channel) plane (16384 blocks).
// ch >= 64: NT copy.  ch < 64: 2-tap segment blend (block-uniform taps,
// zero-coefficient trick for the zero-pad rows -> no divergence).
// ---------------------------------------------------------------------------
__global__ void __launch_bounds__(256)
shift_copy_kernel(const float* __restrict__ x,
                  const int*   __restrict__ sampIdx,
                  const float* __restrict__ sampA,
                  float*       __restrict__ out) {
    int bid = blockIdx.x;            // 0..16383
    int n  = bid >> 8;
    int ch = bid & 255;
    size_t base = ((size_t)n * CTOT + ch) * HW;
    v4f* dst = (v4f*)(out + base);

    if (ch >= NFOLD) {
        // pass-through channels: single-use -> non-temporal both ways
        const v4f* src = (const v4f*)(x + base);
        for (int i = threadIdx.x; i < HW4; i += 256) {
            v4f v = __builtin_nontemporal_load(&src[i]);
            __builtin_nontemporal_store(v, &dst[i]);
        }
    } else {
        int b = n >> 3, s = n & 7;
        int part = ch >> 4;
        int p  = (b * 4 + part) * 8 + s;
        int o0 = sampIdx[2 * p + 0];
        int o1 = sampIdx[2 * p + 1];
        float a0 = sampA[2 * p + 0];
        float a1 = sampA[2 * p + 1];
        bool v0 = (o0 >= 1) & (o0 <= 8);
        bool v1 = (o1 >= 1) & (o1 <= 8);
        int n0 = v0 ? (b * 8 + o0 - 1) : n;    // safe dummy when pad row (zeros)
        int n1 = v1 ? (b * 8 + o1 - 1) : n;
        if (!v0) a0 = 0.f;
        if (!v1) a1 = 0.f;
        const v4f* s0 = (const v4f*)(x + ((size_t)n0 * CTOT + ch) * HW);
        const v4f* s1 = (const v4f*)(x + ((size_t)n1 * CTOT + ch) * HW);
        for (int i = threadIdx.x; i < HW4; i += 256) {
            v4f d0 = s0[i];                    // regular loads: L2 reuse across taps
            v4f d1 = s1[i];
            v4f r  = a0 * d0 + a1 * d1;
            __builtin_nontemporal_store(r, &dst[i]);
        }
    }
}

// ---------------------------------------------------------------------------
extern "C" void kernel_launch(void* const* d_in, const int* in_sizes, int n_in,
                              void* d_out, int out_size, void* d_ws, size_t ws_size,
                              hipStream_t stream) {
    const float* x    = (const float*)d_in[0];
    const float* cow  = (const float*)d_in[1];
    const float* cob  = (const float*)d_in[2];
    const float* f1w  = (const float*)d_in[3];
    const float* f1b  = (const float*)d_in[4];
    const float* f2w  = (const float*)d_in[5];
    const float* f2b  = (const float*)d_in[6];
    const float* cww  = (const float*)d_in[7];
    const float* cwb  = (const float*)d_in[8];

    float* out_x   = (float*)d_out;            // (64,256,56,56)
    float* out_off = out_x + NOUT0;            // (8,1,2)
    float* out_wt  = out_off + 16;             // (8,8,2)

    float* pooled  = (float*)d_ws;                                   // 4096 f
    int*   sampIdx = (int*)((char*)d_ws + 4096 * sizeof(float));     // 512 i32
    float* sampA   = (float*)((char*)d_ws + (4096 + 512) * 4);       // 512 f

    pool_kernel<<<4096, 256, 0, stream>>>(x, pooled);
    head_kernel<<<1, 32, 0, stream>>>(pooled, cow, cob, f1w, f1b, f2w, f2b,
                                      cww, cwb, out_off, out_wt, sampIdx, sampA);
    shift_copy_kernel<<<16384, 256, 0, stream>>>(x, sampIdx, sampA, out_x);
}